// GraphTransFormer_652835029256
// MI455X (gfx1250) — compile-verified
//
#include <hip/hip_runtime.h>
#include <hip/hip_bf16.h>

// ---------------------------------------------------------------------------
// GraphTransformer (TransformerConv x3) for MI455X / gfx1250, wave32.
// * Dense GEMMs: native f32 WMMA (V_WMMA_F32_16X16X4_F32), one wave per
//   16-row x full-N strip -> the big activation matrix streams from HBM once.
// * Weights are pre-packed on device into (k,k+1)-interleaved pairs so each
//   WMMA B-fragment is ONE aligned ds_load_b64 (conflict-free: 32 lanes x 8B
//   = 64 banks), then DMA-staged into LDS with the Tensor Data Mover
//   (tensor_load_to_lds + s_wait_tensorcnt), one 1-D descriptor per block.
// * Edge gather/scatter: 1 wave per edge, lane = 8 contiguous channels
//   (float4 x2 -> 1KB coalesced per wave); per-head reduction via shfl_xor
//   in 8-lane groups; segment softmax via ordered-uint atomicMax + f32
//   atomicAdd in the L2 atomic units (node tables are L2-resident: 20.5MB).
// ---------------------------------------------------------------------------

typedef float    v2f  __attribute__((ext_vector_type(2)));
typedef float    v8f  __attribute__((ext_vector_type(8)));
typedef unsigned u32x4 __attribute__((ext_vector_type(4)));
typedef int      i32x4 __attribute__((ext_vector_type(4)));
typedef int      i32x8 __attribute__((ext_vector_type(8)));

static constexpr int NN       = 20000;
static constexpr int EE       = 256000;
static constexpr int NODE_DIM = 128;
static constexpr int EDGE_DIM = 8;
static constexpr int HID      = 64;
static constexpr int HEADS    = 4;
static constexpr int HC       = HEADS * HID;   // 256
static constexpr int LAYERS   = 3;
static constexpr int CLASSES  = 3;
static constexpr int GRAPHS   = 64;

#if defined(__has_builtin)
#if __has_builtin(__builtin_amdgcn_tensor_load_to_lds) && \
    __has_builtin(__builtin_amdgcn_s_wait_tensorcnt)
#define USE_TDM 1
#endif
#endif

// ordered-uint encoding so atomicMax(uint) == float max
__device__ __forceinline__ unsigned f2ord(float f) {
    unsigned u = __float_as_uint(f);
    return (u & 0x80000000u) ? ~u : (u | 0x80000000u);
}
__device__ __forceinline__ float ord2f(unsigned u) {
    return __uint_as_float((u & 0x80000000u) ? (u & 0x7FFFFFFFu) : ~u);
}
static constexpr unsigned ORD_NEG_INF = 0x007FFFFFu; // f2ord(-inf)

// ---------------------------------------------------------------------------
// Pack W[K,NC] row-major -> Wp pair-interleaved: Wp[(k/2)*2*NC + n*2 + (k&1)]
// so a WMMA B-fragment (W[k][n], W[k+1][n]) is 8B-contiguous.
// ---------------------------------------------------------------------------
__global__ void pack_w_kernel(const float* __restrict__ W,
                              float* __restrict__ Wp, int K, int NC) {
    const int t = blockIdx.x * blockDim.x + threadIdx.x;
    if (t >= K * NC) return;
    const int k = t / NC;
    const int n = t - k * NC;
    Wp[(size_t)((k >> 1) * NC + n) * 2 + (k & 1)] = W[t];
}

// ---------------------------------------------------------------------------
// Stage packed weights (KNC floats, flat) into LDS via the Tensor Data Mover.
// D# per CDNA5 ISA 8.3/8.4: 1-D tile, data_size=4B, tile_dim0=tensor_dim0=KNC.
// Issued by wave 0 only (TDM ignores EXEC); completion via TENSORcnt.
// ---------------------------------------------------------------------------
template <int KNC>
__device__ __forceinline__ void stage_weights(const float* __restrict__ Wp,
                                              float* __restrict__ sW) {
#ifdef USE_TDM
    if (threadIdx.x < 32) {
        const unsigned long long ga = (unsigned long long)(uintptr_t)Wp;
        const unsigned lds = (unsigned)(uintptr_t)sW;   // low 32b = LDS offset
        u32x4 g0 = { 1u,                                // count=1, user mode
                     lds,
                     (unsigned)(ga & 0xFFFFFFFFu),
                     (unsigned)((ga >> 32) & 0x01FFFFFFu) | 0x80000000u }; // type=2
        i32x8 g1 = { 0x00020000,                        // data_size = 4B
                     (int)((KNC & 0xFFFF) << 16),       // tensor_dim0 lo
                     (int)((KNC >> 16) | (1 << 16)),    // tensor_dim0 hi | tensor_dim1=1
                     (int)(KNC << 16),                  // tile_dim0 = KNC (<=16384)
                     1,                                 // tile_dim1 = 1
                     KNC,                               // tensor_dim0_stride lo
                     0, 0 };
        i32x4 z4 = { 0, 0, 0, 0 };
#if __clang_major__ >= 23
        i32x8 z8 = { 0, 0, 0, 0, 0, 0, 0, 0 };
        __builtin_amdgcn_tensor_load_to_lds(g0, g1, z4, z4, z8, 0);
#else
        __builtin_amdgcn_tensor_load_to_lds(g0, g1, z4, z4, 0);
#endif
        __builtin_amdgcn_s_wait_tensorcnt(0);
    }
#else
    for (int i = threadIdx.x; i < KNC; i += blockDim.x) sW[i] = Wp[i];
#endif
}

// ---------------------------------------------------------------------------
// WMMA f32 GEMM:  C[M,NC] = A[M,K] @ W[K,NC] (+bias) (+relu)
// One wave per 16-row strip, NC/16 accumulators (v8f). A-frag (16x4 f32):
// lanes 0-15 rows @ K+0/1, lanes 16-31 same rows @ K+2/3. B-frag = one
// ds_load_b64 from the pair-packed LDS image. M is a multiple of 16 here.
// ---------------------------------------------------------------------------
template <int K, int NC, bool RELU, bool BIAS>
__global__ void gemm_wmma_f32(const float* __restrict__ A,
                              const float* __restrict__ Wp,   // pair-packed
                              const float* __restrict__ bias,
                              float* __restrict__ C, int M) {
    constexpr int NT = NC / 16;
    __shared__ __align__(16) float sW[K * NC];

    stage_weights<K * NC>(Wp, sW);
    __syncthreads();                         // all waves reach this barrier

    const int tile = blockIdx.x * (blockDim.x >> 5) + (threadIdx.x >> 5);
    const int lane = threadIdx.x & 31;
    if (tile * 16 >= M) return;              // wave-uniform; EXEC all-ones for WMMA
    const int rowBase = tile * 16;
    const int half = lane >> 4;              // 0: K+0/1, 1: K+2/3
    const int l16  = lane & 15;

    v8f zero = {};
    v8f acc[NT];
#pragma unroll
    for (int t = 0; t < NT; ++t) acc[t] = zero;

    const float* Arow = A + (size_t)(rowBase + l16) * K;
    __builtin_prefetch(Arow, 0, 3);          // global_prefetch_b8: warm A strip

    for (int kk = 0; kk < K; kk += 4) {
        v2f a;
        a.x = Arow[kk + 2 * half + 0];
        a.y = Arow[kk + 2 * half + 1];
        const int p = (kk >> 1) + half;      // packed pair-row
        const float* sWb = sW + ((size_t)p * NC + l16) * 2;
#pragma unroll
        for (int t = 0; t < NT; ++t) {
            v2f b = *(const v2f*)(sWb + t * 32);   // one ds_load_b64
            acc[t] = __builtin_amdgcn_wmma_f32_16x16x4_f32(
                false, a, false, b, (short)0, acc[t], false, false);
        }
    }
    // D layout: VGPR r -> row rowBase + r + 8*half, col = t*16 + l16
#pragma unroll
    for (int t = 0; t < NT; ++t) {
        const int col = t * 16 + l16;
        float bv = BIAS ? bias[col] : 0.0f;
#pragma unroll
        for (int r = 0; r < 8; ++r) {
            const int row = rowBase + r + 8 * half;
            float v = acc[t][r] + bv;
            if (RELU) v = fmaxf(v, 0.0f);
            C[(size_t)row * NC + col] = v;
        }
    }
}

// ---------------------------------------------------------------------------
// Edge attention logits: alpha[e,h] = scale * sum_c q[dst,h,c]*(k[src,h,c]+ee[e,h,c])
// plus atomic segment-max into amax[dst,h]. One wave per edge, lane owns 8
// contiguous channels (lanes 0-7 -> head0, ..., 24-31 -> head3).
// ---------------------------------------------------------------------------
__global__ void edge_alpha_kernel(const float* __restrict__ q,
                                  const float* __restrict__ k,
                                  const float* __restrict__ ee,
                                  const int* __restrict__ eidx,
                                  float* __restrict__ alpha,
                                  unsigned* __restrict__ amax) {
    const int wave = (blockIdx.x * blockDim.x + threadIdx.x) >> 5;
    const int lane = threadIdx.x & 31;
    if (wave >= EE) return;
    const int src = eidx[wave];
    const int dst = eidx[EE + wave];
    const size_t hc0 = (size_t)lane * 8;

    const float4* qp = (const float4*)(q  + (size_t)dst * HC + hc0);
    const float4* kp = (const float4*)(k  + (size_t)src * HC + hc0);
    const float4* ep = (const float4*)(ee + (size_t)wave * HC + hc0);
    float4 qa = qp[0], qb = qp[1];
    float4 ka = kp[0], kb = kp[1];
    float4 ea = ep[0], eb = ep[1];

    float s = qa.x * (ka.x + ea.x) + qa.y * (ka.y + ea.y) +
              qa.z * (ka.z + ea.z) + qa.w * (ka.w + ea.w) +
              qb.x * (kb.x + eb.x) + qb.y * (kb.y + eb.y) +
              qb.z * (kb.z + eb.z) + qb.w * (kb.w + eb.w);
    // reduce across 8-lane head groups
    s += __shfl_xor(s, 1, 32);
    s += __shfl_xor(s, 2, 32);
    s += __shfl_xor(s, 4, 32);
    if ((lane & 7) == 0) {
        const int hd = lane >> 3;
        const float a = s * 0.125f;          // 1/sqrt(64)
        alpha[(size_t)wave * HEADS + hd] = a;
        atomicMax(&amax[(size_t)dst * HEADS + hd], f2ord(a));
    }
}

// ea = exp(alpha - amax[dst]); denom[dst] += ea   (one thread per edge*head)
__global__ void edge_softmax_kernel(const int* __restrict__ eidx,
                                    float* __restrict__ alpha,
                                    const unsigned* __restrict__ amax,
                                    float* __restrict__ denom) {
    const int t = blockIdx.x * blockDim.x + threadIdx.x;
    if (t >= EE * HEADS) return;
    const int e  = t >> 2;
    const int hd = t & 3;
    const int dst = eidx[EE + e];
    const float m = ord2f(amax[(size_t)dst * HEADS + hd]);
    const float ev = __expf(alpha[t] - m);
    alpha[t] = ev;
    atomicAdd(&denom[(size_t)dst * HEADS + hd], ev);
}

// agg[dst,h,c] += (v[src,h,c] + ee[e,h,c]) * (ea / denom[dst,h])
__global__ void edge_msg_kernel(const float* __restrict__ vmat,
                                const float* __restrict__ ee,
                                const int* __restrict__ eidx,
                                const float* __restrict__ alpha,
                                const float* __restrict__ denom,
                                float* __restrict__ agg) {
    const int wave = (blockIdx.x * blockDim.x + threadIdx.x) >> 5;
    const int lane = threadIdx.x & 31;
    if (wave >= EE) return;
    const int src = eidx[wave];
    const int dst = eidx[EE + wave];
    const int hd  = lane >> 3;
    const float coef = alpha[(size_t)wave * HEADS + hd] /
                       denom[(size_t)dst * HEADS + hd];
    const size_t hc0 = (size_t)lane * 8;
    const float4* vp = (const float4*)(vmat + (size_t)src * HC + hc0);
    const float4* ep = (const float4*)(ee + (size_t)wave * HC + hc0);
    float4 va = vp[0], vb = vp[1];
    float4 ea = ep[0], eb = ep[1];
    float* ap = agg + (size_t)dst * HC + hc0;
    atomicAdd(ap + 0, (va.x + ea.x) * coef);
    atomicAdd(ap + 1, (va.y + ea.y) * coef);
    atomicAdd(ap + 2, (va.z + ea.z) * coef);
    atomicAdd(ap + 3, (va.w + ea.w) * coef);
    atomicAdd(ap + 4, (vb.x + eb.x) * coef);
    atomicAdd(ap + 5, (vb.y + eb.y) * coef);
    atomicAdd(ap + 6, (vb.z + eb.z) * coef);
    atomicAdd(ap + 7, (vb.w + eb.w) * coef);
}

// h_out = relu(LN(mean_heads(agg) + skip + res)) ; one 64-thread block per node
__global__ void node_update_kernel(const float* __restrict__ agg,
                                   const float* __restrict__ skip,
                                   const float* __restrict__ res,
                                   const float* __restrict__ ln_g,
                                   const float* __restrict__ ln_b,
                                   float* __restrict__ hout) {
    const int n = blockIdx.x;
    const int c = threadIdx.x;            // 0..63
    const float* ag = agg + (size_t)n * HC;
    float x = 0.25f * (ag[c] + ag[64 + c] + ag[128 + c] + ag[192 + c]);
    x += skip[(size_t)n * HID + c] + res[(size_t)n * HID + c];

    __shared__ float red[HID];
    red[c] = x;
    __syncthreads();
    for (int s = 32; s > 0; s >>= 1) {
        if (c < s) red[c] += red[c + s];
        __syncthreads();
    }
    const float mean = red[0] * (1.0f / HID);
    __syncthreads();
    const float d = x - mean;
    red[c] = d * d;
    __syncthreads();
    for (int s = 32; s > 0; s >>= 1) {
        if (c < s) red[c] += red[c + s];
        __syncthreads();
    }
    const float var = red[0] * (1.0f / HID);
    const float y = d * rsqrtf(var + 1e-5f) * ln_g[c] + ln_b[c];
    hout[(size_t)n * HID + c] = fmaxf(y, 0.0f);
}

// global mean pool (sum + count) via atomics
__global__ void pool_kernel(const float* __restrict__ h,
                            const int* __restrict__ batch,
                            float* __restrict__ gsum,
                            float* __restrict__ gcnt) {
    const int t = blockIdx.x * blockDim.x + threadIdx.x;
    if (t >= NN * HID) return;
    const int n = t >> 6;
    const int c = t & 63;
    const int g = batch[n];
    atomicAdd(&gsum[(size_t)g * HID + c], h[t]);
    if (c == 0) atomicAdd(&gcnt[g], 1.0f);
}

// pooled @ cls_W + cls_b -> log_softmax ; one thread per graph
__global__ void cls_kernel(const float* __restrict__ gsum,
                           const float* __restrict__ gcnt,
                           const float* __restrict__ W,
                           const float* __restrict__ b,
                           float* __restrict__ out) {
    const int g = threadIdx.x;
    if (g >= GRAPHS) return;
    const float inv = 1.0f / fmaxf(gcnt[g], 1.0f);
    float z0 = b[0], z1 = b[1], z2 = b[2];
    for (int c = 0; c < HID; ++c) {
        const float p = gsum[(size_t)g * HID + c] * inv;
        z0 += p * W[c * CLASSES + 0];
        z1 += p * W[c * CLASSES + 1];
        z2 += p * W[c * CLASSES + 2];
    }
    const float m = fmaxf(z0, fmaxf(z1, z2));
    const float s = __expf(z0 - m) + __expf(z1 - m) + __expf(z2 - m);
    const float ls = m + logf(s);
    out[g * CLASSES + 0] = z0 - ls;
    out[g * CLASSES + 1] = z1 - ls;
    out[g * CLASSES + 2] = z2 - ls;
}

__global__ void fill_f32(float* p, float v, int n) {
    const int i = blockIdx.x * blockDim.x + threadIdx.x;
    if (i < n) p[i] = v;
}
__global__ void fill_u32(unsigned* p, unsigned v, int n) {
    const int i = blockIdx.x * blockDim.x + threadIdx.x;
    if (i < n) p[i] = v;
}

static inline int gemm_blocks(int M) {  // 8 waves (8 row-tiles) per 256-thread block
    const int tiles = M / 16;
    return (tiles + 7) / 8;
}

extern "C" void kernel_launch(void* const* d_in, const int* in_sizes, int n_in,
                              void* d_out, int out_size, void* d_ws, size_t ws_size,
                              hipStream_t stream) {
    (void)in_sizes; (void)n_in; (void)out_size; (void)ws_size;
    const float* x         = (const float*)d_in[0];
    const float* edge_attr = (const float*)d_in[1];
    const float* node_W    = (const float*)d_in[2];
    const float* node_b    = (const float*)d_in[3];
    const float* e1_W      = (const float*)d_in[4];
    const float* e1_b      = (const float*)d_in[5];
    const float* e2_W      = (const float*)d_in[6];
    const float* e2_b      = (const float*)d_in[7];
    const float* Wq        = (const float*)d_in[8];
    const float* bq        = (const float*)d_in[9];
    const float* Wk        = (const float*)d_in[10];
    const float* bk        = (const float*)d_in[11];
    const float* Wv        = (const float*)d_in[12];
    const float* bv        = (const float*)d_in[13];
    const float* We        = (const float*)d_in[14];
    const float* Wskip     = (const float*)d_in[15];
    const float* bskip     = (const float*)d_in[16];
    const float* ln_g      = (const float*)d_in[17];
    const float* ln_b      = (const float*)d_in[18];
    const float* cls_W     = (const float*)d_in[19];
    const float* cls_b     = (const float*)d_in[20];
    const int*   eidx      = (const int*)d_in[21];
    const int*   batch     = (const int*)d_in[22];
    float* out = (float*)d_out;

    // ---- workspace layout (floats) -------------------------------------
    float* w = (float*)d_ws;
    size_t o = 0;
    float* hA     = w + o; o += (size_t)NN * HID;
    float* hB     = w + o; o += (size_t)NN * HID;
    float* e_feat = w + o; o += (size_t)EE * HID;
    float* big    = w + o; o += (size_t)EE * HC;   // ee; t1 aliases its head
    float* t1     = big;                           // [EE,HID] used only pre-layers
    float* qbuf   = w + o; o += (size_t)NN * HC;
    float* kbuf   = w + o; o += (size_t)NN * HC;
    float* vbuf   = w + o; o += (size_t)NN * HC;
    float* skip   = w + o; o += (size_t)NN * HID;
    float* alpha  = w + o; o += (size_t)EE * HEADS;
    unsigned* amax = (unsigned*)(w + o); o += (size_t)NN * HEADS;
    float* denom  = w + o; o += (size_t)NN * HEADS;
    float* agg    = w + o; o += (size_t)NN * HC;
    float* gsum   = w + o; o += (size_t)GRAPHS * HID;
    float* gcnt   = w + o; o += (size_t)GRAPHS;
    // pair-packed weight images
    float* pNode  = w + o; o += (size_t)NODE_DIM * HID;
    float* pE1    = w + o; o += (size_t)EDGE_DIM * HID;
    float* pE2    = w + o; o += (size_t)HID * HID;
    float* pQ[LAYERS]; float* pK[LAYERS]; float* pV[LAYERS];
    float* pEw[LAYERS]; float* pS[LAYERS];
    for (int l = 0; l < LAYERS; ++l) {
        pQ[l]  = w + o; o += (size_t)HID * HC;
        pK[l]  = w + o; o += (size_t)HID * HC;
        pV[l]  = w + o; o += (size_t)HID * HC;
        pEw[l] = w + o; o += (size_t)HID * HC;
        pS[l]  = w + o; o += (size_t)HID * HID;
    }

    auto pack = [&](const float* W, float* Wp, int K, int NC) {
        pack_w_kernel<<<(K * NC + 255) / 256, 256, 0, stream>>>(W, Wp, K, NC);
    };
    pack(node_W, pNode, NODE_DIM, HID);
    pack(e1_W,   pE1,   EDGE_DIM, HID);
    pack(e2_W,   pE2,   HID,      HID);
    for (int l = 0; l < LAYERS; ++l) {
        const size_t wofs = (size_t)l * HID * HC;
        pack(Wq + wofs, pQ[l],  HID, HC);
        pack(Wk + wofs, pK[l],  HID, HC);
        pack(Wv + wofs, pV[l],  HID, HC);
        pack(We + wofs, pEw[l], HID, HC);
        pack(Wskip + (size_t)l * HID * HID, pS[l], HID, HID);
    }

    // ---- node / edge encoders ------------------------------------------
    gemm_wmma_f32<NODE_DIM, HID, false, true>
        <<<gemm_blocks(NN), 256, 0, stream>>>(x, pNode, node_b, hA, NN);
    gemm_wmma_f32<EDGE_DIM, HID, true, true>
        <<<gemm_blocks(EE), 256, 0, stream>>>(edge_attr, pE1, e1_b, t1, EE);
    gemm_wmma_f32<HID, HID, false, true>
        <<<gemm_blocks(EE), 256, 0, stream>>>(t1, pE2, e2_b, e_feat, EE);

    const int edgeWaveBlocks = EE / 8;                  // 8 waves/block, 1 wave/edge
    const int ehThreads = (EE * HEADS + 255) / 256;

    for (int l = 0; l < LAYERS; ++l) {
        const float* hin  = (l & 1) ? hB : hA;
        float*       hout = (l & 1) ? hA : hB;
        const size_t bofs = (size_t)l * HC;

        gemm_wmma_f32<HID, HC, false, true>
            <<<gemm_blocks(NN), 256, 0, stream>>>(hin, pQ[l], bq + bofs, qbuf, NN);
        gemm_wmma_f32<HID, HC, false, true>
            <<<gemm_blocks(NN), 256, 0, stream>>>(hin, pK[l], bk + bofs, kbuf, NN);
        gemm_wmma_f32<HID, HC, false, true>
            <<<gemm_blocks(NN), 256, 0, stream>>>(hin, pV[l], bv + bofs, vbuf, NN);
        gemm_wmma_f32<HID, HC, false, false>
            <<<gemm_blocks(EE), 256, 0, stream>>>(e_feat, pEw[l], nullptr, big, EE);
        gemm_wmma_f32<HID, HID, false, true>
            <<<gemm_blocks(NN), 256, 0, stream>>>(hin, pS[l],
                                                  bskip + (size_t)l * HID, skip, NN);

        fill_u32<<<(NN * HEADS + 255) / 256, 256, 0, stream>>>(amax, ORD_NEG_INF, NN * HEADS);
        fill_f32<<<(NN * HEADS + 255) / 256, 256, 0, stream>>>(denom, 0.0f, NN * HEADS);
        fill_f32<<<(NN * HC + 255) / 256, 256, 0, stream>>>(agg, 0.0f, NN * HC);

        edge_alpha_kernel<<<edgeWaveBlocks, 256, 0, stream>>>(qbuf, kbuf, big, eidx, alpha, amax);
        edge_softmax_kernel<<<ehThreads, 256, 0, stream>>>(eidx, alpha, amax, denom);
        edge_msg_kernel<<<edgeWaveBlocks, 256, 0, stream>>>(vbuf, big, eidx, alpha, denom, agg);

        node_update_kernel<<<NN, HID, 0, stream>>>(agg, skip, hin,
                                                   ln_g + (size_t)l * HID,
                                                   ln_b + (size_t)l * HID, hout);
    }
    const float* hfin = (LAYERS & 1) ? hB : hA;   // LAYERS=3 -> hB

    fill_f32<<<(GRAPHS * HID + 255) / 256, 256, 0, stream>>>(gsum, 0.0f, GRAPHS * HID);
    fill_f32<<<1, 256, 0, stream>>>(gcnt, 0.0f, GRAPHS);
    pool_kernel<<<(NN * HID + 255) / 256, 256, 0, stream>>>(hfin, batch, gsum, gcnt);
    cls_kernel<<<1, 64, 0, stream>>>(gsum, gcnt, cls_W, cls_b, out);
}